// HardNegativeMiner_50869592655657
// MI455X (gfx1250) — compile-verified
//
#include <hip/hip_runtime.h>
#include <stdint.h>

#define TOPK  5
#define TPB   256
#define CHUNK 1024   // floats per staged chunk = 256 threads * 4 floats
#define NBUF  8      // async pipeline depth (32 KB LDS staging)

// ---- gfx1250 async global->LDS path (probe via __has_builtin) --------------
#if defined(__has_builtin)
#if __has_builtin(__builtin_amdgcn_global_load_async_to_lds_b128)
#define HAVE_ASYNC_LDS 1
#endif
#if __has_builtin(__builtin_amdgcn_s_wait_asynccnt)
#define HAVE_WAIT_ASYNC 1
#endif
#endif

typedef int v4i __attribute__((ext_vector_type(4)));
typedef __attribute__((address_space(1))) v4i as1_v4i;   // global (__device__)
typedef __attribute__((address_space(3))) v4i as3_v4i;   // LDS (__shared__)

__device__ __forceinline__ void async_copy16(const float* g, float* l) {
#ifdef HAVE_ASYNC_LDS
  __builtin_amdgcn_global_load_async_to_lds_b128((as1_v4i*)(uintptr_t)g,
                                                 (as3_v4i*)l,
                                                 /*offset=*/0, /*cpol=*/0);
#else
  float4 v = *(const float4*)g;
  *(float4*)l = v;
#endif
}

#ifdef HAVE_WAIT_ASYNC
#define WAIT_ASYNC(n) do { __builtin_amdgcn_s_wait_asynccnt(n); \
                           asm volatile("" ::: "memory"); } while (0)
#else
#ifdef HAVE_ASYNC_LDS
#define WAIT_ASYNC(n) asm volatile("s_wait_asynccnt %0" :: "n"(n) : "memory")
#else
#define WAIT_ASYNC(n) asm volatile("" ::: "memory")
#endif
#endif

// Monotone key: bigger key == bigger float; ties -> smaller index wins.
__device__ __forceinline__ unsigned long long make_key(float f, int idx) {
  unsigned u = __float_as_uint(f);
  unsigned m = (u & 0x80000000u) ? ~u : (u | 0x80000000u);
  return ((unsigned long long)m << 32) | (unsigned)(~idx);
}

// Direct macro on scalar locals: no address-taken captures -> stays in VGPRs.
#define UPDATE5(f, i)                                                       \
  do {                                                                      \
    if ((i) != label) {                                                     \
      unsigned long long k_ = make_key((f), (i));                           \
      if (k_ > t4) {                                                        \
        if (k_ > t2) {                                                      \
          if (k_ > t1) {                                                    \
            if (k_ > t0) { t4 = t3; t3 = t2; t2 = t1; t1 = t0; t0 = k_; }   \
            else         { t4 = t3; t3 = t2; t2 = t1; t1 = k_; }            \
          } else         { t4 = t3; t3 = t2; t2 = k_; }                     \
        } else {                                                            \
          if (k_ > t3)   { t4 = t3; t3 = k_; }                              \
          else             t4 = k_;                                         \
        }                                                                   \
      }                                                                     \
    }                                                                       \
  } while (0)

__global__ __launch_bounds__(TPB) void
HardNegativeMiner_topk_kernel(const float* __restrict__ logits,
                              const int* __restrict__ labels,
                              int* __restrict__ out, int B, int N) {
  __shared__ unsigned long long smem[NBUF * TPB * 2];  // 32 KB (staging + merge)
  float* stage = (float*)smem;

  const int b   = blockIdx.x;
  const int tid = threadIdx.x;
  const float* row = logits + (size_t)b * (size_t)N;
  const int label  = labels[b];

  unsigned long long t0 = 0, t1 = 0, t2 = 0, t3 = 0, t4 = 0;

  // ---- scalar head until 16B alignment (row base is only 4B aligned) ----
  int head = (int)(((16u - ((unsigned)(uintptr_t)row & 15u)) & 15u) >> 2);
  if (head > N) head = N;
  if (tid < head) { float f = row[tid]; UPDATE5(f, tid); }

  const int nchunk     = (N - head) / CHUNK;
  const int main_iters = (nchunk > (NBUF - 1)) ? nchunk - (NBUF - 1) : 0;

  // ---- prologue: fill the async pipeline ----
  const int pre = (nchunk < NBUF) ? nchunk : NBUF;
  for (int c = 0; c < pre; ++c)
    async_copy16(row + head + c * CHUNK + tid * 4,
                 stage + (((c & (NBUF - 1)) * TPB + tid) << 2));

  // ---- steady state: wait oldest chunk, consume, issue next ----
  int c = 0;
  for (; c < main_iters; ++c) {
    WAIT_ASYNC(NBUF - 1);                              // oldest chunk landed
    const int buf  = c & (NBUF - 1);
    float4 v = *(float4*)(stage + ((buf * TPB + tid) << 2));
    const int base = head + c * CHUNK + tid * 4;
    UPDATE5(v.x, base + 0);
    UPDATE5(v.y, base + 1);
    UPDATE5(v.z, base + 2);
    UPDATE5(v.w, base + 3);
    const int nc = c + NBUF;
    if (nc < nchunk)
      async_copy16(row + head + nc * CHUNK + tid * 4,
                   stage + (((nc & (NBUF - 1)) * TPB + tid) << 2));
  }
  // ---- drain: everything already issued, just wait-all and consume ----
  for (; c < nchunk; ++c) {
    WAIT_ASYNC(0);
    const int buf  = c & (NBUF - 1);
    float4 v = *(float4*)(stage + ((buf * TPB + tid) << 2));
    const int base = head + c * CHUNK + tid * 4;
    UPDATE5(v.x, base + 0);
    UPDATE5(v.y, base + 1);
    UPDATE5(v.z, base + 2);
    UPDATE5(v.w, base + 3);
  }
  WAIT_ASYNC(0);

  // ---- scalar tail ----
  for (int i = head + nchunk * CHUNK + tid; i < N; i += TPB) {
    float f = row[i];
    UPDATE5(f, i);
  }

  // ---- merge 256 * 5 candidates ----
  __syncthreads();
  smem[tid * TOPK + 0] = t0;
  smem[tid * TOPK + 1] = t1;
  smem[tid * TOPK + 2] = t2;
  smem[tid * TOPK + 3] = t3;
  smem[tid * TOPK + 4] = t4;
  __syncthreads();

  if (tid < 32) {
    for (int p = 0; p < TOPK; ++p) {
      unsigned long long lmax = 0; int li = -1;
      for (int i = tid; i < TPB * TOPK; i += 32) {
        unsigned long long k = smem[i];
        if (k > lmax) { lmax = k; li = i; }
      }
      unsigned long long g = lmax;
      for (int off = 16; off > 0; off >>= 1) {
        unsigned long long o = __shfl_xor(g, off, 32);
        if (o > g) g = o;
      }
      unsigned long long ball = __ballot(li >= 0 && lmax == g);
      int first = __ffsll(ball) - 1;
      if (tid == first) {
        smem[li] = 0;                                  // remove winner
        out[b * TOPK + p] = (int)(~(unsigned)(g & 0xFFFFFFFFull));
      }
      __builtin_amdgcn_wave_barrier();
    }
  }
}

extern "C" void kernel_launch(void* const* d_in, const int* in_sizes, int n_in,
                              void* d_out, int out_size, void* d_ws, size_t ws_size,
                              hipStream_t stream) {
  (void)n_in; (void)d_ws; (void)ws_size; (void)out_size;
  const float* logits = (const float*)d_in[0];
  const int*   labels = (const int*)d_in[1];
  int*         out    = (int*)d_out;

  int B = in_sizes[1];
  if (B <= 0) B = 8192;
  int N = in_sizes[0] / B;
  if (N <= 0) N = 50257;

  HardNegativeMiner_topk_kernel<<<B, TPB, 0, stream>>>(logits, labels, out, B, N);
}